// FourierLinear_15375982920474
// MI455X (gfx1250) — compile-verified
//
#include <hip/hip_runtime.h>

// FNO spectral block as a chain of pruned-DFT GEMMs on fp32 WMMA (gfx1250).
// Pipeline: tables -> zero(Y2) -> G1(rfft-x) -> G2(fft-y) -> G3(mode mix)
//           -> G4(ifft-y) -> G5(irfft-x, normalized).
// fx axis padded 17 -> 32 so all addressing is shift-only and stores are
// unconditional; LDS staging uses GLOBAL_LOAD_ASYNC_TO_LDS_B128 when the
// toolchain exposes it (ASYNCcnt path), else coalesced float4 copies.

typedef __attribute__((ext_vector_type(2))) float v2f;
typedef __attribute__((ext_vector_type(8))) float v8f;

__device__ __forceinline__ v8f wmma4(v2f a, v2f b, v8f c) {
  // D = A(16x4,f32) * B(4x16,f32) + C(16x16,f32)
  return __builtin_amdgcn_wmma_f32_16x16x4_f32(
      false, a, false, b, (short)0, c, false, false);
}

#ifdef __has_builtin
#if __has_builtin(__builtin_amdgcn_global_load_async_to_lds_b128)
#define USE_ASYNC_LDS 1
#endif
#endif
#ifndef USE_ASYNC_LDS
#define USE_ASYNC_LDS 0
#endif

#if USE_ASYNC_LDS
typedef int v4i_t __attribute__((vector_size(16)));
typedef __attribute__((address_space(1))) v4i_t* async_gptr_t;
typedef __attribute__((address_space(3))) v4i_t* async_lptr_t;
#define ASYNC_CP16(gsrc, ldst)                                           \
  __builtin_amdgcn_global_load_async_to_lds_b128(                        \
      (async_gptr_t)(gsrc), (async_lptr_t)(ldst), 0, 0)
#define ASYNC_WAIT() asm volatile("s_wait_asynccnt 0x0" ::: "memory")
#endif

#define NP   256   // pixels per axis
#define NC   32    // channels in/out
#define NB   16    // batch
#define FX   17    // kept x-frequencies (0..16)
#define FXP  32    // fx stride padded to a power of two
#define FYP  48    // kept y-frequency rows (33) padded to 3 tiles of 16
#define TWO_PI_OVER_256 0.024543692606170259f

// ---------------------------------------------------------------- tables ----
__global__ void k_tables(float* __restrict__ T1c, float* __restrict__ T1s,
                         float* __restrict__ F2re, float* __restrict__ F2im,
                         float* __restrict__ Ere,  float* __restrict__ Eim,
                         float* __restrict__ Pre,  float* __restrict__ Pim) {
  int t = blockIdx.x * blockDim.x + threadIdx.x;
  if (t < 8192) {                       // T1[32][256]: forward x DFT, e^{-i th}
    int m = t >> 8, xx = t & 255;
    float c = 0.f, s = 0.f;
    if (m <= 16) { int ph = (m * xx) & 255; float ss, cc;
      __sincosf(TWO_PI_OVER_256 * (float)ph, &ss, &cc); c = cc; s = -ss; }
    T1c[t] = c; T1s[t] = s;
  } else if (t < 20480) {               // F2[48][256]: forward y DFT rows
    int u = t - 8192; int r = u >> 8, y = u & 255;
    float c = 0.f, s = 0.f;
    if (r < 33) { int fy = (r <= 16) ? r : (r + 223);
      int ph = (fy * y) & 255; float ss, cc;
      __sincosf(TWO_PI_OVER_256 * (float)ph, &ss, &cc); c = cc; s = -ss; }
    F2re[u] = c; F2im[u] = s;
  } else if (t < 32768) {               // E[256][48]: inverse y, e^{+i th}
    int u = t - 20480; int y = u / FYP, r = u % FYP;
    float c = 0.f, s = 0.f;
    if (r < 33) { int fy = (r <= 16) ? r : (r + 223);
      int ph = (fy * y) & 255;
      __sincosf(TWO_PI_OVER_256 * (float)ph, &s, &c); }
    Ere[u] = c; Eim[u] = s;
  } else if (t < 40960) {               // P[256][32]: c2r x, scaled 1/65536
    int u = t - 32768; int xx = u >> 5, k = u & 31;
    float c = 0.f, s = 0.f;
    if (k <= 16) {
      float sc = ((k == 0) ? 1.0f : 2.0f) / 65536.0f;
      int ph = (k * xx) & 255; float ss, cc;
      __sincosf(TWO_PI_OVER_256 * (float)ph, &ss, &cc);
      c = sc * cc; s = sc * ss;
    }
    Pre[u] = c; Pim[u] = s;
  }
}

__global__ void k_zero(float* __restrict__ p, int n) {
  int t = blockIdx.x * blockDim.x + threadIdx.x;
  if (t < n) p[t] = 0.f;
}

// ------------------------------------------------ G1: rfft along x (r2c) ----
// grid = NB*NP (one block per (b,y)); block = 256 (8 waves).
// Out: X1[b][y][fxp(32)][c], complex.  A = T1 (32x256, rows>16 zero), B = data.
__global__ void k_fft_x(const float* __restrict__ x,
                        float* __restrict__ X1re, float* __restrict__ X1im,
                        const float* __restrict__ T1c,
                        const float* __restrict__ T1s) {
  __shared__ float lds[NP * NC];                       // 32 KB: [x][c]
  const int by = blockIdx.x;                           // b*256 + y
  const float* src = x + (size_t)by * NP * NC;
#if USE_ASYNC_LDS
  for (int t = threadIdx.x; t < (NP * NC) / 4; t += blockDim.x)
    ASYNC_CP16(src + (t << 2), lds + (t << 2));
  ASYNC_WAIT();
#else
  for (int t = threadIdx.x; t < (NP * NC) / 4; t += blockDim.x)
    ((float4*)lds)[t] = ((const float4*)src)[t];
#endif
  __syncthreads();

  const int w     = threadIdx.x >> 5;
  const int lane  = threadIdx.x & 31;
  const int l15   = lane & 15;
  const int lh    = (lane >> 4) & 1;
  const int reim  = w & 1;
  const int mtile = (w >> 1) & 1;
  const int ntile = (w >> 2) & 1;
  const float* T  = reim ? T1s : T1c;
  const int m = mtile * 16 + l15;                      // fx row of table
  const int n = ntile * 16 + l15;                      // channel col

  v8f acc = {};
  for (int k0 = 0; k0 < NP; k0 += 4) {
    const int kb = k0 + 2 * lh;
    v2f a; a.x = T[m * NP + kb];       a.y = T[m * NP + kb + 1];
    v2f b; b.x = lds[kb * NC + n];     b.y = lds[(kb + 1) * NC + n];
    acc = wmma4(a, b, acc);
  }
  float* out = reim ? X1im : X1re;
  const size_t obase = ((size_t)by * FXP) * NC + n;
  for (int j = 0; j < 8; ++j) {
    int fx = mtile * 16 + j + 8 * lh;                  // 0..31, pad rows are 0
    out[obase + (size_t)fx * NC] = acc[j];
  }
}

// ---------------------------------------------- G2: full fft along y (c2c) --
// grid = NB*FX (one block per (b,fx)); block = 192 (6 waves: 3 mtile x 2 ntile)
// Out: X2[b][r(48)][fxp][c] complex.  K=256 over y, staged in 2 LDS chunks.
__global__ void k_fft_y(const float* __restrict__ X1re,
                        const float* __restrict__ X1im,
                        float* __restrict__ X2re, float* __restrict__ X2im,
                        const float* __restrict__ F2re,
                        const float* __restrict__ F2im) {
  __shared__ float lre[128 * NC];                      // 16 KB
  __shared__ float lim[128 * NC];                      // 16 KB
  const int bc = blockIdx.x;
  const int b = bc / FX, c = bc % FX;

  const int lane  = threadIdx.x & 31;
  const int l15   = lane & 15;
  const int lh    = (lane >> 4) & 1;
  const int w     = threadIdx.x >> 5;
  const int mtile = w % 3;
  const int ntile = w / 3;
  const int m = mtile * 16 + l15;                      // fy row (0..47)
  const int n = ntile * 16 + l15;                      // channel

  v8f P = {}, Q = {}, R = {}, S = {};
  for (int ch = 0; ch < 2; ++ch) {
    __syncthreads();
#if USE_ASYNC_LDS
    for (int t = threadIdx.x; t < 128 * (NC / 4); t += blockDim.x) {
      int y = (t >> 3) + ch * 128, q = t & 7;
      size_t rb = ((size_t)(b * NP + y) * FXP + c) * NC + (q << 2);
      ASYNC_CP16(X1re + rb, lre + (t << 2));
      ASYNC_CP16(X1im + rb, lim + (t << 2));
    }
    ASYNC_WAIT();
#else
    for (int t = threadIdx.x; t < 128 * (NC / 4); t += blockDim.x) {
      int y = (t >> 3) + ch * 128, q = t & 7;
      size_t rb = (((size_t)(b * NP + y) * FXP + c) * NC) >> 2;
      ((float4*)lre)[t] = ((const float4*)X1re)[rb + q];
      ((float4*)lim)[t] = ((const float4*)X1im)[rb + q];
    }
#endif
    __syncthreads();
    for (int k0 = 0; k0 < 128; k0 += 4) {
      const int kb = k0 + 2 * lh;
      const int kg = ch * 128 + kb;                    // global y index
      v2f ar; ar.x = F2re[m * NP + kg];    ar.y = F2re[m * NP + kg + 1];
      v2f ai; ai.x = F2im[m * NP + kg];    ai.y = F2im[m * NP + kg + 1];
      v2f br; br.x = lre[kb * NC + n];     br.y = lre[(kb + 1) * NC + n];
      v2f bi; bi.x = lim[kb * NC + n];     bi.y = lim[(kb + 1) * NC + n];
      P = wmma4(ar, br, P);  Q = wmma4(ai, bi, Q);
      R = wmma4(ar, bi, R);  S = wmma4(ai, br, S);
    }
  }
  const size_t ob = (((size_t)b * FYP) * FXP + c) * NC + n;
  for (int j = 0; j < 8; ++j) {
    int r = mtile * 16 + j + 8 * lh;                   // 0..47 (rows>=33 are 0)
    size_t o = ob + (size_t)r * FXP * NC;
    X2re[o] = P[j] - Q[j];
    X2im[o] = R[j] + S[j];
  }
}

// -------------------------------------- G3: per-mode complex channel mix ----
// grid = m modes; block = 64 (2 waves: ntile). M=16 batch, K=32 in, N=32 out.
__global__ void k_mix(const float* __restrict__ X2re,
                      const float* __restrict__ X2im,
                      const float* __restrict__ w_re,
                      const float* __restrict__ w_im,
                      const int* __restrict__ idx0,
                      const int* __restrict__ idx1,
                      float* __restrict__ Y2re, float* __restrict__ Y2im) {
  const int f = blockIdx.x;
  const int i0 = idx0[f];
  const int c  = idx1[f];
  const int r  = (i0 <= 16) ? i0 : (i0 - 223);         // row in padded fy space

  const int lane  = threadIdx.x & 31;
  const int l15   = lane & 15;                          // batch for A-frag
  const int lh    = (lane >> 4) & 1;
  const int ntile = threadIdx.x >> 5;
  const int n = ntile * 16 + l15;                       // output channel

  const size_t abase = (((size_t)l15 * FYP + r) * FXP + c) * NC;
  const size_t wbase = (size_t)f * NC * NC;

  v8f P = {}, Q = {}, R = {}, S = {};
  for (int k0 = 0; k0 < NC; k0 += 4) {
    const int kb = k0 + 2 * lh;
    v2f ar; ar.x = X2re[abase + kb];  ar.y = X2re[abase + kb + 1];
    v2f ai; ai.x = X2im[abase + kb];  ai.y = X2im[abase + kb + 1];
    v2f br; br.x = w_re[wbase + kb * NC + n];  br.y = w_re[wbase + (kb + 1) * NC + n];
    v2f bi; bi.x = w_im[wbase + kb * NC + n];  bi.y = w_im[wbase + (kb + 1) * NC + n];
    P = wmma4(ar, br, P);  Q = wmma4(ai, bi, Q);
    R = wmma4(ar, bi, R);  S = wmma4(ai, br, S);
  }
  const size_t ob = ((size_t)r * FXP + c) * NC + n;
  for (int j = 0; j < 8; ++j) {
    int bb = j + 8 * lh;                                // batch
    size_t o = ob + (size_t)bb * FYP * FXP * NC;
    Y2re[o] = P[j] - Q[j];
    Y2im[o] = R[j] + S[j];
  }
}

// ------------------------------------------- G4: inverse fft along y (c2c) --
// grid = NB*FX*16 (one block per (b,fx,ytile)); block = 64 (2 waves: ntile).
__global__ void k_ifft_y(const float* __restrict__ Y2re,
                         const float* __restrict__ Y2im,
                         float* __restrict__ Y1re, float* __restrict__ Y1im,
                         const float* __restrict__ Ere,
                         const float* __restrict__ Eim) {
  const int gid = blockIdx.x;
  const int mt = gid & 15;
  const int bc = gid >> 4;
  const int b = bc / FX, c = bc % FX;

  const int lane  = threadIdx.x & 31;
  const int l15   = lane & 15;
  const int lh    = (lane >> 4) & 1;
  const int ntile = threadIdx.x >> 5;
  const int m = mt * 16 + l15;                          // output y row
  const int n = ntile * 16 + l15;                       // channel

  v8f P = {}, Q = {}, R = {}, S = {};
  for (int k0 = 0; k0 < FYP; k0 += 4) {
    const int kb = k0 + 2 * lh;
    v2f ar; ar.x = Ere[m * FYP + kb];  ar.y = Ere[m * FYP + kb + 1];
    v2f ai; ai.x = Eim[m * FYP + kb];  ai.y = Eim[m * FYP + kb + 1];
    size_t b0 = (((size_t)b * FYP + kb) * FXP + c) * NC + n;
    size_t b1 = b0 + (size_t)FXP * NC;
    __builtin_prefetch(&Y2re[b0 + 4 * FXP * NC], 0, 0);
    v2f br; br.x = Y2re[b0];  br.y = Y2re[b1];
    v2f bi; bi.x = Y2im[b0];  bi.y = Y2im[b1];
    P = wmma4(ar, br, P);  Q = wmma4(ai, bi, Q);
    R = wmma4(ar, bi, R);  S = wmma4(ai, br, S);
  }
  const size_t ob = (((size_t)b * NP) * FXP + c) * NC + n;
  for (int j = 0; j < 8; ++j) {
    int y = mt * 16 + j + 8 * lh;
    size_t o = ob + (size_t)y * FXP * NC;
    Y1re[o] = P[j] - Q[j];
    Y1im[o] = R[j] + S[j];
  }
}

// ----------------------------- G5: inverse rfft along x (c2r, normalized) --
// grid = NB*NP (one block per (b,y)); block = 256 (8 waves: 4 mgroups x 2 nt).
__global__ void k_ifft_x(const float* __restrict__ Y1re,
                         const float* __restrict__ Y1im,
                         float* __restrict__ out,
                         const float* __restrict__ Pre,
                         const float* __restrict__ Pim) {
  __shared__ float lre[32 * NC];                        // 4 KB, k padded to 32
  __shared__ float lim[32 * NC];
  const int by = blockIdx.x;                            // b*256 + y
  for (int t = threadIdx.x; t < 32 * NC; t += blockDim.x) {
    int k = t >> 5, i = t & 31;
    float vr = 0.f, vi = 0.f;
    if (k < FX) {
      size_t o = ((size_t)by * FXP + k) * NC + i;
      vr = Y1re[o]; vi = Y1im[o];
    }
    lre[t] = vr; lim[t] = vi;
  }
  __syncthreads();

  const int w    = threadIdx.x >> 5;
  const int lane = threadIdx.x & 31;
  const int l15  = lane & 15;
  const int lh   = (lane >> 4) & 1;
  const int ntile = w & 1;
  const int mg    = w >> 1;                             // 0..3, 4 mtiles each
  const int n = ntile * 16 + l15;                       // output channel

  for (int s = 0; s < 4; ++s) {
    const int mt = mg * 4 + s;
    const int m = mt * 16 + l15;                        // output x
    v8f P = {}, Q = {};
    for (int k0 = 0; k0 < 32; k0 += 4) {
      const int kb = k0 + 2 * lh;
      v2f ar; ar.x = Pre[m * 32 + kb];   ar.y = Pre[m * 32 + kb + 1];
      v2f ai; ai.x = Pim[m * 32 + kb];   ai.y = Pim[m * 32 + kb + 1];
      v2f br; br.x = lre[kb * NC + n];   br.y = lre[(kb + 1) * NC + n];
      v2f bi; bi.x = lim[kb * NC + n];   bi.y = lim[(kb + 1) * NC + n];
      P = wmma4(ar, br, P);
      Q = wmma4(ai, bi, Q);
    }
    for (int j = 0; j < 8; ++j) {
      int xx = mt * 16 + j + 8 * lh;
      out[((size_t)by * NP + xx) * NC + n] = P[j] - Q[j];
    }
  }
}

// --------------------------------------------------------------- launcher --
extern "C" void kernel_launch(void* const* d_in, const int* in_sizes, int n_in,
                              void* d_out, int out_size, void* d_ws, size_t ws_size,
                              hipStream_t stream) {
  const float* x    = (const float*)d_in[0];
  const float* w_re = (const float*)d_in[1];
  const float* w_im = (const float*)d_in[2];
  const int*   idx0 = (const int*)d_in[3];
  const int*   idx1 = (const int*)d_in[4];
  const int m_modes = in_sizes[3];

  float* ws = (float*)d_ws;
  const size_t n1 = (size_t)NB * NP * FXP * NC;   // 4,194,304 floats
  const size_t n2 = (size_t)NB * FYP * FXP * NC;  //   786,432 floats

  float* X1re = ws;            float* X1im = X1re + n1;
  float* X2re = X1im + n1;     float* X2im = X2re + n2;
  float* Y2re = X2im + n2;     float* Y2im = Y2re + n2;
  // Y1 aliases X1: X1 is fully consumed by k_fft_y before k_ifft_y writes Y1.
  float* Y1re = X1re;          float* Y1im = X1im;
  float* T1c  = Y2im + n2;     float* T1s  = T1c + 8192;
  float* F2re = T1s + 8192;    float* F2im = F2re + 12288;
  float* Ere  = F2im + 12288;  float* Eim  = Ere + 12288;
  float* Pre  = Eim + 12288;   float* Pim  = Pre + 8192;

  k_tables<<<160, 256, 0, stream>>>(T1c, T1s, F2re, F2im, Ere, Eim, Pre, Pim);

  const int nz = (int)(2 * n2);                  // Y2re+Y2im contiguous
  k_zero<<<(nz + 255) / 256, 256, 0, stream>>>(Y2re, nz);

  k_fft_x <<<NB * NP,       256, 0, stream>>>(x, X1re, X1im, T1c, T1s);
  k_fft_y <<<NB * FX,       192, 0, stream>>>(X1re, X1im, X2re, X2im, F2re, F2im);
  k_mix   <<<m_modes,        64, 0, stream>>>(X2re, X2im, w_re, w_im, idx0, idx1,
                                              Y2re, Y2im);
  k_ifft_y<<<NB * FX * 16,   64, 0, stream>>>(Y2re, Y2im, Y1re, Y1im, Ere, Eim);
  k_ifft_x<<<NB * NP,       256, 0, stream>>>(Y1re, Y1im, (float*)d_out, Pre, Pim);
}